// HierarchicalSoftmaxLoss_36679020708606
// MI455X (gfx1250) — compile-verified
//
#include <hip/hip_runtime.h>

// Hierarchical softmax loss (depth-2, B=128) for MI455X / gfx1250.
// Per sample: loss = lse(root[0:128]) - root[parent]
//                  + lse(seg[parent]) - seg[parent][child],  mean over batch.
// Memory-bound (~4.2 MB useful traffic); Sum-of-exp done on the matrix pipe
// via V_WMMA_F32_16X16X4_F32 with an all-ones A operand (co-executes with
// the VALU exp work). Deterministic two-kernel reduction (no float atomics).

#define BF 128
#define PRED_SIZE (BF + BF * BF)   // 16512
#define BATCH 4096
#define PITCH 132                  // LDS row pitch (floats); 132 % 64 banks -> conflict-free
#define SPW 16                     // samples per wave (one WMMA tile of columns)
#define WAVES 4                    // waves per block
#define THREADS (WAVES * 32)

typedef __attribute__((ext_vector_type(2))) float v2f;
typedef __attribute__((ext_vector_type(8))) float v8f;

__device__ __forceinline__ float wave_max_bcast(float v) {
#pragma unroll
  for (int m = 16; m >= 1; m >>= 1)
    v = fmaxf(v, __shfl_xor(v, m, 32));
  return v;
}

// Process one 16-sample x 128-logit segment for this wave.
// LEVEL=0: root segment (offset 0). LEVEL=1: parent-owned segment.
// Returns, in every lane, logsumexp of sample (lane & 15)'s segment.
template <int LEVEL>
__device__ __forceinline__ float segment_lse(const float* __restrict__ pred,
                                             const int* __restrict__ targets,
                                             int sbase, float* __restrict__ slice,
                                             int lane) {
  const int n = lane & 15;
  float mymax = 0.0f;

  // Phase A: coalesced B128 row loads; row-max via 5 shuffle-xor steps;
  // exp(x - max) staged to LDS in the wave's private slice.
#pragma unroll 4
  for (int m = 0; m < SPW; ++m) {
    int off = 0;
    if (LEVEL) {
      int t = targets[sbase + m];          // uniform -> scalar load
      off = BF + ((t >> 7) << 7);
    }
    const float4* rp =
        (const float4*)(pred + (size_t)(sbase + m) * PRED_SIZE + off);
    float4 v = rp[lane];
    float rm = fmaxf(fmaxf(v.x, v.y), fmaxf(v.z, v.w));
    rm = wave_max_bcast(rm);
    if (n == m) mymax = rm;                // park sample m's max in lane m (both halves)
    float4 e;
    e.x = __expf(v.x - rm);
    e.y = __expf(v.y - rm);
    e.z = __expf(v.z - rm);
    e.w = __expf(v.w - rm);
    *(float4*)(slice + m * PITCH + 4 * lane) = e;
  }
  __syncthreads();

  // Phase B: row sums on the matrix pipe. A = ones(16x4), B = 4x16 exp tile,
  // accumulate 32 K-steps: D[m][n] = sum_{j<128} exp[n][j] for every m.
  v8f acc = {};
  v2f ones;
  ones[0] = 1.0f;
  ones[1] = 1.0f;
  const int h2 = (lane >> 4) << 1;         // K half owned by upper lanes
  const float* bp = slice + n * PITCH + h2;
#pragma unroll 8
  for (int k = 0; k < BF / 4; ++k) {
    v2f b;
    b[0] = bp[k * 4 + 0];                  // merges to ds_load_b64, conflict-free
    b[1] = bp[k * 4 + 1];
    acc = __builtin_amdgcn_wmma_f32_16x16x4_f32(
        false, ones, false, b, (short)0, acc, false, false);
  }
  __syncthreads();                         // WAR: slice is reused by next segment

  return mymax + __logf(acc[0]);           // acc[0] = D[row][n] = sum_exp(sample n)
}

__global__ void __launch_bounds__(THREADS)
hsm_loss_kernel(const float* __restrict__ pred, const int* __restrict__ targets,
                float* __restrict__ losses) {
  __shared__ float lds[WAVES * SPW * PITCH];   // 33 KB
  const int lane = threadIdx.x & 31;
  const int wave = threadIdx.x >> 5;
  const int sbase = (blockIdx.x * WAVES + wave) * SPW;
  float* slice = lds + wave * SPW * PITCH;

  float lse1 = segment_lse<0>(pred, targets, sbase, slice, lane);
  float lse2 = segment_lse<1>(pred, targets, sbase, slice, lane);

  if (lane < SPW) {
    const int s = sbase + lane;
    const int t = targets[s];
    const int parent = t >> 7;
    const int child = t & (BF - 1);
    const float* row = pred + (size_t)s * PRED_SIZE;
    float logit1 = row[parent];
    float logit2 = row[BF + (parent << 7) + child];
    losses[s] = (lse1 - logit1) + (lse2 - logit2);
  }
}

__global__ void __launch_bounds__(256)
reduce_mean_kernel(const float* __restrict__ losses, float* __restrict__ out) {
  __shared__ float sm[256];
  float s = 0.0f;
  for (int i = threadIdx.x; i < BATCH; i += 256) s += losses[i];
  sm[threadIdx.x] = s;
  __syncthreads();
#pragma unroll
  for (int off = 128; off > 0; off >>= 1) {
    if ((int)threadIdx.x < off) sm[threadIdx.x] += sm[threadIdx.x + off];
    __syncthreads();
  }
  if (threadIdx.x == 0) out[0] = sm[0] * (1.0f / BATCH);
}

extern "C" void kernel_launch(void* const* d_in, const int* in_sizes, int n_in,
                              void* d_out, int out_size, void* d_ws, size_t ws_size,
                              hipStream_t stream) {
  const float* pred = (const float*)d_in[0];   // [4096, 16512] f32
  const int* targets = (const int*)d_in[1];    // [4096] i32
  float* losses = (float*)d_ws;                // 4096 f32 scratch
  float* out = (float*)d_out;                  // scalar f32

  const int blocks = BATCH / (WAVES * SPW);    // 64
  hsm_loss_kernel<<<blocks, THREADS, 0, stream>>>(pred, targets, losses);
  reduce_mean_kernel<<<1, 256, 0, stream>>>(losses, out);
}